// PatchCorrAttention_2095944040859
// MI455X (gfx1250) — compile-verified
//
#include <hip/hip_runtime.h>
#include <math.h>

typedef __attribute__((ext_vector_type(2))) float v2f;
typedef __attribute__((ext_vector_type(8))) float v8f;

#define CHN   64
#define CRN   32
#define NPIX  1600     // 40*40
#define NPIXD 400      // 20*20
#define L1V   1444     // 38*38 (base patches / queries)
#define LTOT  1768     // 1444 + 18*18 (all keys)
#define LPAD  1776     // 111 * 16
#define NKT   111      // key tiles
#define NQT   91       // query tiles (1456 >= 1444)
#define KKP   16       // padded patch-dim stride (9 -> 16)

// workspace layout (float offsets)
#define O_PARTS   0
#define O_WPARTS  256
#define O_SCALES  320
#define O_BASE    384
#define O_WF      (O_BASE + CRN*NPIX)
#define O_KQ      (O_WF + CRN*NPIXD)
#define O_VT      (O_KQ + CRN*LPAD*KKP)
#define O_OUTP    (O_VT + CRN*KKP*LPAD)
#define O_OIMG    (O_OUTP + CRN*9*L1V)

// cross-half lane swap (lane ^ 16) as a pure-VALU op: v_permlanex16_b32 with
// identity selects. Avoids ds_bpermute + s_wait_dscnt in the attention hot loop.
__device__ __forceinline__ float xor16(float v) {
#if __has_builtin(__builtin_amdgcn_permlanex16)
  int i = __builtin_bit_cast(int, v);
  int r = __builtin_amdgcn_permlanex16(i, i, 0x76543210, 0xFEDCBA98, false, false);
  return __builtin_bit_cast(float, r);
#else
  return __shfl_xor(v, 16, 32);
#endif
}

// ---------- stage 1: base = PReLU(w1 x + b1), partial sum of squares ----------
__global__ __launch_bounds__(256) void base_kernel(
    const float* __restrict__ x, const float* __restrict__ w1,
    const float* __restrict__ b1, const float* __restrict__ a1p,
    float* __restrict__ base, float* __restrict__ parts) {
  int idx = blockIdx.x * 256 + threadIdx.x;            // < 51200 exactly
  int c = idx / NPIX, p = idx % NPIX;
  float acc = b1[c];
  const float* wr = w1 + c * CHN;
#pragma unroll 8
  for (int ch = 0; ch < CHN; ++ch) acc += wr[ch] * x[ch * NPIX + p];
  float a = *a1p;
  float val = acc >= 0.f ? acc : a * acc;
  base[idx] = val;
  float sq = val * val;
#pragma unroll
  for (int m = 1; m < 32; m <<= 1) sq += __shfl_xor(sq, m, 32);
  __shared__ float red[8];
  int lane = threadIdx.x & 31, wid = threadIdx.x >> 5;
  if (lane == 0) red[wid] = sq;
  __syncthreads();
  if (threadIdx.x == 0) {
    float t = 0.f;
#pragma unroll
    for (int i = 0; i < 8; ++i) t += red[i];
    parts[blockIdx.x] = t;
  }
}

// ---------- stage 2: wf = PReLU(w2 avgpool2(x) + b2), partial sumsq ----------
__global__ __launch_bounds__(256) void wf_kernel(
    const float* __restrict__ x, const float* __restrict__ w2,
    const float* __restrict__ b2, const float* __restrict__ a2p,
    float* __restrict__ wf, float* __restrict__ parts) {
  int idx = blockIdx.x * 256 + threadIdx.x;            // < 12800 exactly
  int c = idx / NPIXD, p2 = idx % NPIXD;
  int i2 = p2 / 20, j2 = p2 % 20;
  float acc = b2[c];
  const float* wr = w2 + c * CHN;
#pragma unroll 4
  for (int ch = 0; ch < CHN; ++ch) {
    const float* xb = x + ch * NPIX + (2 * i2) * 40 + 2 * j2;
    float xv = 0.25f * (xb[0] + xb[1] + xb[40] + xb[41]);
    acc += wr[ch] * xv;
  }
  float a = *a2p;
  float val = acc >= 0.f ? acc : a * acc;
  wf[idx] = val;
  float sq = val * val;
#pragma unroll
  for (int m = 1; m < 32; m <<= 1) sq += __shfl_xor(sq, m, 32);
  __shared__ float red[8];
  int lane = threadIdx.x & 31, wid = threadIdx.x >> 5;
  if (lane == 0) red[wid] = sq;
  __syncthreads();
  if (threadIdx.x == 0) {
    float t = 0.f;
#pragma unroll
    for (int i = 0; i < 8; ++i) t += red[i];
    parts[blockIdx.x] = t;
  }
}

// ---------- stage 3: deterministic final reduction -> normalization scales ----------
__global__ __launch_bounds__(256) void scales_kernel(
    const float* __restrict__ parts, const float* __restrict__ wparts,
    float* __restrict__ scales) {
  __shared__ float sh[256];
  float t = 0.f;
  for (int i = threadIdx.x; i < 200; i += 256) t += parts[i];
  sh[threadIdx.x] = t;
  __syncthreads();
  for (int s = 128; s > 0; s >>= 1) {
    if (threadIdx.x < s) sh[threadIdx.x] += sh[threadIdx.x + s];
    __syncthreads();
  }
  float bs = sh[0];
  __syncthreads();
  t = 0.f;
  for (int i = threadIdx.x; i < 50; i += 256) t += wparts[i];
  sh[threadIdx.x] = t;
  __syncthreads();
  for (int s = 128; s > 0; s >>= 1) {
    if (threadIdx.x < s) sh[threadIdx.x] += sh[threadIdx.x + s];
    __syncthreads();
  }
  if (threadIdx.x == 0) {
    float sq10 = sqrtf(10.0f);                 // sqrt(softmax scale), folded into Q and K
    scales[0] = sq10 * rsqrtf(bs);             // base branch: sqrt(10)/base_max
    scales[1] = sq10 * rsqrtf(sh[0]);          // wf branch:   sqrt(10)/w_max
  }
}

// ---------- stage 4: pack unfolded patches into WMMA-friendly buffers ----------
// Kq[c][l][kk]  (kk stride 16, kk>=9 and l>=LTOT zero) = pan * sqrt(10)
// VT[c][kk][l]  (unnormalized values, transposed)
__global__ __launch_bounds__(256) void pack_kernel(
    const float* __restrict__ base, const float* __restrict__ wf,
    const float* __restrict__ scales, float* __restrict__ Kq,
    float* __restrict__ VT) {
  int idx = blockIdx.x * 256 + threadIdx.x;            // < 32*1776 exactly
  int c = idx / LPAD, l = idx % LPAD;
  float pav[9];
  float sc = 0.f;
  if (l < L1V) {
    int i = l / 38, j = l % 38;
    const float* bp = base + c * NPIX + i * 40 + j;
#pragma unroll
    for (int di = 0; di < 3; ++di)
#pragma unroll
      for (int dj = 0; dj < 3; ++dj) pav[di * 3 + dj] = bp[di * 40 + dj];
    sc = scales[0];
  } else if (l < LTOT) {
    int l2 = l - L1V;
    int i = l2 / 18, j = l2 % 18;
    const float* wp = wf + c * NPIXD + i * 20 + j;
#pragma unroll
    for (int di = 0; di < 3; ++di)
#pragma unroll
      for (int dj = 0; dj < 3; ++dj) pav[di * 3 + dj] = wp[di * 20 + dj];
    sc = scales[1];
  } else {
#pragma unroll
    for (int k = 0; k < 9; ++k) pav[k] = 0.f;
  }
  float* kqr = Kq + ((size_t)c * LPAD + l) * KKP;
#pragma unroll
  for (int kk = 0; kk < KKP; ++kk) {
    float pv = (kk < 9) ? pav[kk] : 0.f;
    kqr[kk] = pv * sc;
    VT[((size_t)c * KKP + kk) * LPAD + l] = pv;
  }
}

// ---------- stage 5: per-channel flash attention, 16 queries per wave ----------
// Sᵀ = K_tile × Qᵀ  (3 × v_wmma_f32_16x16x4_f32) : rows=keys, cols=queries
// online softmax over keys (8 in-lane ops + 1 v_permlanex16 per reduction)
// outᵀ += Vᵀ × Pᵀ   (4 × v_wmma_f32_16x16x4_f32)
__global__ __launch_bounds__(32) void attn_kernel(
    const float* __restrict__ Kq, const float* __restrict__ VT,
    float* __restrict__ outp) {
  const int c = blockIdx.y;
  const int qt = blockIdx.x;
  const int lane = threadIdx.x;
  const int lm = lane & 15;
  const int h = lane >> 4;                       // lane half
  const float* Kc = Kq + (size_t)c * LPAD * KKP;
  const float* VTc = VT + (size_t)c * KKP * LPAD;
  const int qbase = qt * 16;

  // B operand (Qᵀ) chunks j=0..2: lane needs Q[q=lm][4j+2h], [4j+2h+1]
  v2f bq[3];
#pragma unroll
  for (int j = 0; j < 3; ++j) {
    const float* p = Kc + (size_t)(qbase + lm) * KKP + (4 * j + 2 * h);
    bq[j].x = p[0];
    bq[j].y = p[1];
  }

  v8f acc = {};              // outᵀ accumulator: row v = r+8h, col q = lm
  float mrow = -1e30f;
  float lsum = 0.f;

  for (int kt = 0; kt < NKT; ++kt) {
    const int kbase = kt * 16;
    // ---- scores: Sᵀ tile ----
    v8f s = {};
#pragma unroll
    for (int j = 0; j < 3; ++j) {
      const float* p = Kc + (size_t)(kbase + lm) * KKP + (4 * j + 2 * h);
      v2f a;
      a.x = p[0];
      a.y = p[1];
      s = __builtin_amdgcn_wmma_f32_16x16x4_f32(false, a, false, bq[j],
                                                (short)0, s, false, false);
    }
    // ---- online softmax over keys (rows) ----
    float tm = s[0];
#pragma unroll
    for (int r = 1; r < 8; ++r) tm = fmaxf(tm, s[r]);
    tm = fmaxf(tm, xor16(tm));
    float mnew = fmaxf(mrow, tm);
    float alpha = __expf(mrow - mnew);
    mrow = mnew;
    float tsum = 0.f;
#pragma unroll
    for (int r = 0; r < 8; ++r) {
      int key = kbase + r + 8 * h;
      float pv = (key < LTOT) ? __expf(s[r] - mnew) : 0.f;  // mask padded keys
      s[r] = pv;
      tsum += pv;
    }
    tsum += xor16(tsum);
    lsum = lsum * alpha + tsum;
#pragma unroll
    for (int r = 0; r < 8; ++r) acc[r] *= alpha;
    // ---- reshape Pᵀ (D layout) -> B operand chunks via half-swap + select ----
    float ssw[8];
#pragma unroll
    for (int r = 0; r < 8; ++r) ssw[r] = xor16(s[r]);
    v2f b2[4];
    b2[0].x = h ? ssw[2] : s[0];
    b2[0].y = h ? ssw[3] : s[1];
    b2[1].x = h ? ssw[6] : s[4];
    b2[1].y = h ? ssw[7] : s[5];
    b2[2].x = h ? s[2] : ssw[0];
    b2[2].y = h ? s[3] : ssw[1];
    b2[3].x = h ? s[6] : ssw[4];
    b2[3].y = h ? s[7] : ssw[5];
    // ---- accumulate outᵀ += Vᵀ × Pᵀ ----
#pragma unroll
    for (int j = 0; j < 4; ++j) {
      const float* p = VTc + (size_t)lm * LPAD + kbase + 4 * j + 2 * h;
      v2f a;
      a.x = p[0];
      a.y = p[1];
      acc = __builtin_amdgcn_wmma_f32_16x16x4_f32(false, a, false, b2[j],
                                                  (short)0, acc, false, false);
    }
  }

  // ---- normalize + store (only real queries, only 9 value rows) ----
  float inv = 1.0f / lsum;
  int q = qbase + lm;
  if (q < L1V) {
#pragma unroll
    for (int r = 0; r < 8; ++r) {
      int v = r + 8 * h;
      if (v < 9) outp[((size_t)c * 9 + v) * L1V + q] = acc[r] * inv;
    }
  }
}

// ---------- stage 6: fold (sum overlapping patches) / 9 ----------
__global__ __launch_bounds__(256) void fold_kernel(
    const float* __restrict__ outp, float* __restrict__ oimg) {
  int idx = blockIdx.x * 256 + threadIdx.x;            // < 51200 exactly
  int c = idx / NPIX, p = idx % NPIX;
  int i = p / 40, j = p % 40;
  float sum = 0.f;
#pragma unroll
  for (int di = 0; di < 3; ++di) {
    int ii = i - di;
    if (ii < 0 || ii >= 38) continue;
#pragma unroll
    for (int dj = 0; dj < 3; ++dj) {
      int jj = j - dj;
      if (jj < 0 || jj >= 38) continue;
      sum += outp[((size_t)c * 9 + di * 3 + dj) * L1V + ii * 38 + jj];
    }
  }
  oimg[idx] = sum * (1.0f / 9.0f);
}

// ---------- stage 7: 1x1 conv (w3) + PReLU + residual ----------
__global__ __launch_bounds__(256) void final_kernel(
    const float* __restrict__ oimg, const float* __restrict__ x,
    const float* __restrict__ w3, const float* __restrict__ b3,
    const float* __restrict__ a3p, float* __restrict__ out) {
  int idx = blockIdx.x * 256 + threadIdx.x;            // < 102400 exactly
  int o = idx / NPIX, p = idx % NPIX;
  float acc = b3[o];
  const float* wr = w3 + o * CRN;
#pragma unroll 8
  for (int c = 0; c < CRN; ++c) acc += wr[c] * oimg[c * NPIX + p];
  float a = *a3p;
  float y = acc >= 0.f ? acc : a * acc;
  out[idx] = y + x[idx];
}

extern "C" void kernel_launch(void* const* d_in, const int* in_sizes, int n_in,
                              void* d_out, int out_size, void* d_ws, size_t ws_size,
                              hipStream_t stream) {
  const float* x  = (const float*)d_in[0];
  const float* w1 = (const float*)d_in[1];
  const float* b1 = (const float*)d_in[2];
  const float* a1 = (const float*)d_in[3];
  const float* w2 = (const float*)d_in[4];
  const float* b2 = (const float*)d_in[5];
  const float* a2 = (const float*)d_in[6];
  const float* w3 = (const float*)d_in[7];
  const float* b3 = (const float*)d_in[8];
  const float* a3 = (const float*)d_in[9];
  float* out = (float*)d_out;
  float* ws = (float*)d_ws;

  base_kernel<<<200, 256, 0, stream>>>(x, w1, b1, a1, ws + O_BASE, ws + O_PARTS);
  wf_kernel<<<50, 256, 0, stream>>>(x, w2, b2, a2, ws + O_WF, ws + O_WPARTS);
  scales_kernel<<<1, 256, 0, stream>>>(ws + O_PARTS, ws + O_WPARTS, ws + O_SCALES);
  pack_kernel<<<(CRN * LPAD) / 256, 256, 0, stream>>>(ws + O_BASE, ws + O_WF,
                                                      ws + O_SCALES, ws + O_KQ,
                                                      ws + O_VT);
  dim3 grid(NQT, CRN);
  attn_kernel<<<grid, 32, 0, stream>>>(ws + O_KQ, ws + O_VT, ws + O_OUTP);
  fold_kernel<<<200, 256, 0, stream>>>(ws + O_OUTP, ws + O_OIMG);
  final_kernel<<<400, 256, 0, stream>>>(ws + O_OIMG, x, w3, b3, a3, out);
}